// MeshGenLoss_55319178772943
// MI455X (gfx1250) — compile-verified
//
#include <hip/hip_runtime.h>

typedef __attribute__((ext_vector_type(2))) float v2f;
typedef __attribute__((ext_vector_type(8))) float v8f;

#define BIGV 1e6f
#define CHUNK 1024

// Row-min of pairwise squared distances between X[b] (rows) and Y[b] (cols).
// One wave per 16-row strip. Homogeneous-coordinate WMMA:
//   A row  = (-2x0, -2x1, -2x2, 1)        (16x4 f32 A layout)
//   B col  = ( y0,   y1,   y2,  |y|^2)    (4x16 f32 B layout)
//   C init = |x|^2 per row
// => V_WMMA_F32_16X16X4_F32 directly emits the 16x16 tile of squared
// distances; the epilogue is pure min-accumulation.
// Y is staged through LDS pre-swizzled per lane-half so the B operand is a
// single ds_load_b64. SELF=1 adds BIG on the diagonal.
template <int SELF>
__global__ __launch_bounds__(256) void rowmin_wmma(
    const float* __restrict__ X, const float* __restrict__ Y,
    float* __restrict__ out, int N)
{
    // plane 0: (y0,y1) for lanes 0-15 ; plane 1: (y2,|y|^2) for lanes 16-31
    __shared__ float2 sy[2][CHUNK];

    const int tid  = threadIdx.x;
    const int lane = tid & 31;
    const int wave = tid >> 5;
    const int half = lane >> 4;
    const int l16  = lane & 15;

    const int blocksPerBatch = N / 128;           // 8 waves * 16 rows per block
    const int b            = blockIdx.x / blocksPerBatch;
    const int blockInBatch = blockIdx.x % blocksPerBatch;
    const int ibase        = (blockInBatch * 8 + wave) * 16;

    // A tile (loop-invariant): lanes 0-15 hold K={0,1}, lanes 16-31 K={2,3}
    const float* xp = X + ((size_t)b * N + (ibase + l16)) * 3;
    float ax = xp[0], ay = xp[1], az = xp[2];
    float myx2 = ax * ax + ay * ay + az * az;
    v2f a;
    a[0] = half ? -2.f * az : -2.f * ax;
    a[1] = half ? 1.f       : -2.f * ay;

    // C init: |x|^2 of the 8 rows this lane's accumulators cover
    // (C/D layout: VGPR v -> row v + 8*half; row r held in-half at lane r%16).
    v8f cinit;
#pragma unroll
    for (int v = 0; v < 8; ++v)
        cinit[v] = __shfl(myx2, v + 24 * half, 32);

    float racc[8];
#pragma unroll
    for (int v = 0; v < 8; ++v) racc[v] = 3.4e38f;

    const float* yb = Y + (size_t)b * N * 3;

    for (int chunk = 0; chunk < N; chunk += CHUNK) {
        __syncthreads();
        for (int idx = tid; idx < CHUNK; idx += 256) {
            const float* qp = yb + (size_t)(chunk + idx) * 3;
            float qx = qp[0], qy = qp[1], qz = qp[2];
            sy[0][idx] = make_float2(qx, qy);
            sy[1][idx] = make_float2(qz, qx * qx + qy * qy + qz * qz);
        }
        __syncthreads();

#pragma unroll 2
        for (int jt = 0; jt < CHUNK / 16; ++jt) {
            float2 q = sy[half][jt * 16 + l16];
            v2f bv;
            bv[0] = q.x;
            bv[1] = q.y;

            v8f d = __builtin_amdgcn_wmma_f32_16x16x4_f32(
                        false, a, false, bv, (short)0, cinit, false, false);

            if (SELF) {
                const int jcol = chunk + jt * 16 + l16;
#pragma unroll
                for (int v = 0; v < 8; ++v) {
                    float dd = d[v];
                    if ((ibase + v + 8 * half) == jcol) dd += BIGV;
                    racc[v] = fminf(racc[v], dd);
                }
            } else {
#pragma unroll
                for (int v = 0; v < 8; ++v)
                    racc[v] = fminf(racc[v], d[v]);
            }
        }
    }

    // Finish row-min: min across the 16 lanes of each half (xor stays in-half).
#pragma unroll
    for (int v = 0; v < 8; ++v) {
        float r = racc[v];
#pragma unroll
        for (int m = 1; m < 16; m <<= 1)
            r = fminf(r, __shfl_xor(r, m, 32));
        racc[v] = r;
    }
    if (l16 == 0) {
#pragma unroll
        for (int v = 0; v < 8; ++v)
            out[(size_t)b * N + ibase + v + 8 * half] = racc[v];
    }
}

// Single-block finalize: chamfer mean, per-batch ddof=1 std of NN dists,
// KL term, sizing MSE -> scalar loss.
__global__ __launch_bounds__(256) void finalize_kernel(
    const float* __restrict__ rm0, const float* __restrict__ rm1,
    const float* __restrict__ rm2,
    const float* __restrict__ ps,  const float* __restrict__ ts,
    const float* __restrict__ mu,  const float* __restrict__ lv,
    float* __restrict__ out, int B, int N, int L)
{
    __shared__ float red[256];
    const int tid = threadIdx.x;
    const int BN = B * N, BL = B * L;

    float s01 = 0.f, ssz = 0.f, skl = 0.f;
    float sb[8], qb[8];
#pragma unroll
    for (int i = 0; i < 8; ++i) { sb[i] = 0.f; qb[i] = 0.f; }

    for (int i = tid; i < BN; i += 256) {
        s01 += rm0[i] + rm1[i];
        float r = rm2[i];
        int bb = i / N;
        sb[bb] += r;
        qb[bb] += r * r;
        float dsz = ps[i] - ts[i];
        ssz += dsz * dsz;
    }
    for (int i = tid; i < BL; i += 256) {
        float m = mu[i], l = lv[i];
        skl += 1.f + l - m * m - expf(l);
    }

    auto bsum = [&](float v) -> float {
        red[tid] = v; __syncthreads();
        for (int s = 128; s > 0; s >>= 1) {
            if (tid < s) red[tid] += red[tid + s];
            __syncthreads();
        }
        float r = red[0]; __syncthreads();
        return r;
    };

    float S01 = bsum(s01);
    float SSZ = bsum(ssz);
    float SKL = bsum(skl);
    float dens = 0.f;
    for (int bb = 0; bb < B; ++bb) {
        float SB = bsum(sb[bb]);
        float QB = bsum(qb[bb]);
        float var = (QB - SB * SB / (float)N) / (float)(N - 1);
        dens += sqrtf(fmaxf(var, 0.f));
    }

    if (tid == 0) {
        float cd      = S01 / (float)BN;
        float density = dens / (float)B;
        float sizing  = SSZ / (float)BN;
        float kl      = -0.5f * SKL / (float)BL;
        out[0] = cd + 0.001f * kl + 0.1f * density + 0.05f * sizing;
    }
}

extern "C" void kernel_launch(void* const* d_in, const int* in_sizes, int n_in,
                              void* d_out, int out_size, void* d_ws, size_t ws_size,
                              hipStream_t stream)
{
    const float* pred_pos      = (const float*)d_in[0];
    const float* pred_sizing   = (const float*)d_in[1];
    const float* target_pos    = (const float*)d_in[2];
    const float* target_sizing = (const float*)d_in[3];
    const float* mu            = (const float*)d_in[4];
    const float* lv            = (const float*)d_in[5];

    const int B = 4;
    const int N = in_sizes[1] / B;   // 4096
    const int L = in_sizes[4] / B;   // 512

    float* rm0 = (float*)d_ws;
    float* rm1 = rm0 + (size_t)B * N;
    float* rm2 = rm1 + (size_t)B * N;

    dim3 grid(B * (N / 128)), block(256);
    rowmin_wmma<0><<<grid, block, 0, stream>>>(pred_pos,   target_pos, rm0, N);
    rowmin_wmma<0><<<grid, block, 0, stream>>>(target_pos, pred_pos,   rm1, N);
    rowmin_wmma<1><<<grid, block, 0, stream>>>(pred_pos,   pred_pos,   rm2, N);
    finalize_kernel<<<1, 256, 0, stream>>>(rm0, rm1, rm2,
                                           pred_sizing, target_sizing,
                                           mu, lv, (float*)d_out, B, N, L);
}